// Sinkhorn_8830452761358
// MI455X (gfx1250) — compile-verified
//
#include <hip/hip_runtime.h>
#include <math.h>

// Sinkhorn + Adam for MI455X (gfx1250, wave32).
// 10 iterations x (row-logsumexp kernel, col-logsumexp kernel) + final exp kernel.
// - sum-of-exp reductions run on the matrix pipe via V_WMMA_F32_16X16X4_F32 with a
//   ones operand (16 row/col sums per instruction), co-executing with v_exp_f32.
// - pass-1 computes x = la/tau + dual and caches it in LDS (padded stride, conflict
//   free); pass-2 replays from LDS -> halves L2 read traffic per phase.
// - 64MB matrix is fully L2-resident (192MB L2); HBM floor is ~one 64MB read + one
//   64MB write for the final plan.

#define BATCH 16
#define NDIM 1024
#define MDIM 1024
#define XSTRIDE 1028   // 1028 mod 64 == 4 -> wave b64 access covers all 64 LDS banks

constexpr float INV_TAU  = 20.0f;     // 1 / 0.05
constexpr float LR       = 0.5f;
constexpr float BETA1    = 0.9f;
constexpr float BETA2    = 0.999f;
constexpr float EPS_ADAM = 1e-8f;
constexpr float NEG_BIG  = -3.0e38f;

typedef float v2f __attribute__((ext_vector_type(2)));
typedef float v4f __attribute__((ext_vector_type(4)));
typedef float v8f __attribute__((ext_vector_type(8)));

constexpr size_t XBUF_BYTES = (size_t)16 * XSTRIDE * sizeof(float);  // 65792 B dynamic LDS

// ---------------------------------------------------------------------------
// Row phase: for each (b, n): lse = logsumexp_m(la*INV_TAU + v[m]); Adam update u.
// Grid: 16 batches * 64 row-blocks (16 rows) = 1024 blocks, 128 threads (4 waves).
// Wave w covers columns [w*256, w*256+256). Lane L: row = L&15, K-half h = L>>4
// (16x4 f32 A-matrix layout: lanes 0-15 hold K={0,1}, lanes 16-31 hold K={2,3}).
// ---------------------------------------------------------------------------
__global__ __launch_bounds__(128) void sinkhorn_row(
    const float* __restrict__ la, const float* __restrict__ log_a,
    float* __restrict__ u, const float* __restrict__ v,
    float* __restrict__ mu, float* __restrict__ vu, float bc1, float bc2)
{
  extern __shared__ float xs[];            // [16][XSTRIDE] cached x values
  __shared__ v2f part[4][16];

  const int blk  = blockIdx.x;
  const int b    = blk >> 6;
  const int rb   = blk & 63;
  const int tid  = threadIdx.x;
  const int wave = tid >> 5;
  const int lane = tid & 31;
  const int r    = lane & 15;
  const int h    = lane >> 4;
  const int row  = rb * 16 + r;
  const float* __restrict__ rowp = la + ((size_t)b * NDIM + row) * MDIM;
  const float* __restrict__ vb   = v + b * MDIM;
  float* __restrict__ xr = xs + r * XSTRIDE;
  const int c0 = wave * 256;

  // Pass 1: compute x = la*INV_TAU + v, cache in LDS, track wave-local max.
  float mx0 = NEG_BIG, mx1 = NEG_BIG;
  #pragma unroll 4
  for (int c = c0; c < c0 + 256; c += 8) {
    const int cc = c + h * 2;
    v2f xa = *(const v2f*)(rowp + cc);
    v2f va = *(const v2f*)(vb + cc);
    v2f xb = *(const v2f*)(rowp + cc + 4);
    v2f vb2 = *(const v2f*)(vb + cc + 4);
    v2f pa, pb;
    pa.x = fmaf(xa.x, INV_TAU, va.x);
    pa.y = fmaf(xa.y, INV_TAU, va.y);
    pb.x = fmaf(xb.x, INV_TAU, vb2.x);
    pb.y = fmaf(xb.y, INV_TAU, vb2.y);
    *(v2f*)(xr + cc)     = pa;
    *(v2f*)(xr + cc + 4) = pb;
    mx0 = fmaxf(mx0, fmaxf(pa.x, pa.y));
    mx1 = fmaxf(mx1, fmaxf(pb.x, pb.y));
  }
  float mx = fmaxf(mx0, mx1);
  mx = fmaxf(mx, __shfl_xor(mx, 16, 32));   // lanes L and L^16 share row L&15

  // Pass 2 (LDS only): sum of exp via WMMA row-sum (B = ones), dual accumulators.
  v2f ones; ones.x = 1.0f; ones.y = 1.0f;
  v8f acc0 = {}; v8f acc1 = {};
  #pragma unroll 2
  for (int c = c0; c < c0 + 256; c += 8) {
    const int cc = c + h * 2;
    v2f pa = *(const v2f*)(xr + cc);
    v2f pb = *(const v2f*)(xr + cc + 4);
    v2f ea, eb;
    ea.x = __expf(pa.x - mx);
    ea.y = __expf(pa.y - mx);
    eb.x = __expf(pb.x - mx);
    eb.y = __expf(pb.y - mx);
    acc0 = __builtin_amdgcn_wmma_f32_16x16x4_f32(false, ea, false, ones, (short)0, acc0, false, false);
    acc1 = __builtin_amdgcn_wmma_f32_16x16x4_f32(false, eb, false, ones, (short)0, acc1, false, false);
  }
  v8f acc = acc0 + acc1;

  // D layout: VGPR j, lanes 0-15 -> M=j ; lanes 16-31 -> M=j+8. Every column equal.
  // Lanes {0..7} own rows 0..7, lanes {16..23} own rows 8..15.
  float s = 0.0f;
  #pragma unroll
  for (int j = 0; j < 8; ++j)
    if ((lane & 7) == j) s = acc[j];
  const int rr = (lane & 7) + h * 8;       // row whose sum this lane holds
  float mxr = __shfl(mx, rr, 32);          // that row's wave-local max (lives at lane rr)

  if ((lane & 8) == 0) { v2f t; t.x = mxr; t.y = s; part[wave][rr] = t; }
  __syncthreads();

  if (tid < 16) {
    float M = NEG_BIG;
    #pragma unroll
    for (int w = 0; w < 4; ++w) M = fmaxf(M, part[w][tid].x);
    float S = 0.0f;
    #pragma unroll
    for (int w = 0; w < 4; ++w) S += part[w][tid].y * __expf(part[w][tid].x - M);
    const int idx = b * NDIM + rb * 16 + tid;
    const float lse = M + __logf(S);
    const float g   = log_a[idx] - lse - u[idx];
    const float m_n = BETA1 * mu[idx] + (1.0f - BETA1) * g;
    const float v_n = BETA2 * vu[idx] + (1.0f - BETA2) * g * g;
    mu[idx] = m_n;
    vu[idx] = v_n;
    u[idx]  = u[idx] + LR * (m_n / bc1) / (sqrtf(v_n / bc2) + EPS_ADAM);
  }
}

// ---------------------------------------------------------------------------
// Col phase: for each (b, m): lse = logsumexp_n(la*INV_TAU + u[n]); Adam update v.
// Grid: 16 batches * 64 col-blocks (16 cols) = 1024 blocks, 128 threads (4 waves).
// Wave w covers rows [w*256, w*256+256). Lane L: col = L&15, K-half h = L>>4.
// WMMA with A = ones gives column sums replicated in every D row (use acc[0]).
// ---------------------------------------------------------------------------
__global__ __launch_bounds__(128) void sinkhorn_col(
    const float* __restrict__ la, const float* __restrict__ log_b,
    const float* __restrict__ u, float* __restrict__ v,
    float* __restrict__ mv, float* __restrict__ vvw, float bc1, float bc2)
{
  extern __shared__ float xs[];            // [16 cols][XSTRIDE rows] cached x values
  __shared__ v2f part[4][16];

  const int blk  = blockIdx.x;
  const int b    = blk >> 6;
  const int cb   = blk & 63;
  const int tid  = threadIdx.x;
  const int wave = tid >> 5;
  const int lane = tid & 31;
  const int q    = lane & 15;
  const int h    = lane >> 4;
  const int col  = cb * 16 + q;
  const float* __restrict__ base = la + (size_t)b * NDIM * MDIM + col;
  const float* __restrict__ ub   = u + b * NDIM;
  float* __restrict__ xq = xs + q * XSTRIDE;
  const int r0 = wave * 256;

  // Pass 1: compute x = la*INV_TAU + u, cache in LDS, track wave-local max.
  float mx0 = NEG_BIG, mx1 = NEG_BIG;
  #pragma unroll 4
  for (int c = r0; c < r0 + 256; c += 8) {
    const int n0 = c + h * 2;
    v2f ua  = *(const v2f*)(ub + n0);
    v2f ub2 = *(const v2f*)(ub + n0 + 4);
    v2f pa, pb;
    pa.x = fmaf(base[(size_t)n0 * MDIM],       INV_TAU, ua.x);
    pa.y = fmaf(base[(size_t)(n0 + 1) * MDIM], INV_TAU, ua.y);
    pb.x = fmaf(base[(size_t)(n0 + 4) * MDIM], INV_TAU, ub2.x);
    pb.y = fmaf(base[(size_t)(n0 + 5) * MDIM], INV_TAU, ub2.y);
    *(v2f*)(xq + n0)     = pa;
    *(v2f*)(xq + n0 + 4) = pb;
    mx0 = fmaxf(mx0, fmaxf(pa.x, pa.y));
    mx1 = fmaxf(mx1, fmaxf(pb.x, pb.y));
  }
  float mx = fmaxf(mx0, mx1);
  mx = fmaxf(mx, __shfl_xor(mx, 16, 32));   // lanes L, L^16 share column L&15

  // Pass 2 (LDS only): column sums via WMMA with A = ones.
  v2f ones; ones.x = 1.0f; ones.y = 1.0f;
  v8f acc0 = {}; v8f acc1 = {};
  #pragma unroll 2
  for (int c = r0; c < r0 + 256; c += 8) {
    const int n0 = c + h * 2;
    v2f pa = *(const v2f*)(xq + n0);
    v2f pb = *(const v2f*)(xq + n0 + 4);
    v2f ea, eb;
    ea.x = __expf(pa.x - mx);
    ea.y = __expf(pa.y - mx);
    eb.x = __expf(pb.x - mx);
    eb.y = __expf(pb.y - mx);
    acc0 = __builtin_amdgcn_wmma_f32_16x16x4_f32(false, ones, false, ea, (short)0, acc0, false, false);
    acc1 = __builtin_amdgcn_wmma_f32_16x16x4_f32(false, ones, false, eb, (short)0, acc1, false, false);
  }
  v8f acc = acc0 + acc1;
  float s = acc[0];   // column sum of column (lane&15), replicated in all D rows

  if (lane < 16) { v2f t; t.x = mx; t.y = s; part[wave][q] = t; }
  __syncthreads();

  if (tid < 16) {
    float M = NEG_BIG;
    #pragma unroll
    for (int w = 0; w < 4; ++w) M = fmaxf(M, part[w][tid].x);
    float S = 0.0f;
    #pragma unroll
    for (int w = 0; w < 4; ++w) S += part[w][tid].y * __expf(part[w][tid].x - M);
    const int idx = b * MDIM + cb * 16 + tid;
    const float lse = M + __logf(S);
    const float g   = log_b[idx] - lse - v[idx];
    const float m_n = BETA1 * mv[idx] + (1.0f - BETA1) * g;
    const float v_n = BETA2 * vvw[idx] + (1.0f - BETA2) * g * g;
    mv[idx] = m_n;
    vvw[idx] = v_n;
    v[idx]  = v[idx] + LR * (m_n / bc1) / (sqrtf(v_n / bc2) + EPS_ADAM);
  }
}

// ---------------------------------------------------------------------------
// Final: out = exp(la*INV_TAU + u[n] + v[m]), float4-vectorized (HBM write bound).
// ---------------------------------------------------------------------------
__global__ __launch_bounds__(256) void sinkhorn_out(
    const float* __restrict__ la, const float* __restrict__ u,
    const float* __restrict__ v, float* __restrict__ out)
{
  const size_t i = ((size_t)blockIdx.x * 256 + threadIdx.x) * 4;
  const int b   = (int)(i >> 20);
  const int rem = (int)(i & ((1u << 20) - 1));
  const int n   = rem >> 10;
  const int m   = rem & 1023;
  v4f x  = *(const v4f*)(la + i);
  v4f vv = *(const v4f*)(v + b * MDIM + m);
  const float uu = u[b * NDIM + n];
  v4f o;
  o.x = __expf(fmaf(x.x, INV_TAU, uu + vv.x));
  o.y = __expf(fmaf(x.y, INV_TAU, uu + vv.y));
  o.z = __expf(fmaf(x.z, INV_TAU, uu + vv.z));
  o.w = __expf(fmaf(x.w, INV_TAU, uu + vv.w));
  *(v4f*)(out + i) = o;
}

// ---------------------------------------------------------------------------
extern "C" void kernel_launch(void* const* d_in, const int* in_sizes, int n_in,
                              void* d_out, int out_size, void* d_ws, size_t ws_size,
                              hipStream_t stream) {
  (void)in_sizes; (void)n_in; (void)out_size; (void)ws_size;
  const float* la    = (const float*)d_in[0];   // log_alpha [16,1024,1024]
  const float* log_a = (const float*)d_in[1];   // [16,1024]
  const float* log_b = (const float*)d_in[2];   // [16,1024]

  // Workspace: 6 * 16*1024 floats = 384 KB (u, v, m_u, v_u, m_v, v_v), zeroed each call.
  float* ws = (float*)d_ws;
  float* u   = ws;
  float* v   = ws + 1 * BATCH * NDIM;
  float* m_u = ws + 2 * BATCH * NDIM;
  float* v_u = ws + 3 * BATCH * NDIM;
  float* m_v = ws + 4 * BATCH * NDIM;
  float* v_v = ws + 5 * BATCH * NDIM;
  hipMemsetAsync(d_ws, 0, 6 * BATCH * NDIM * sizeof(float), stream);

  float b1t = 1.0f, b2t = 1.0f;
  for (int it = 0; it < 10; ++it) {
    b1t *= BETA1;
    b2t *= BETA2;
    const float bc1 = 1.0f - b1t;
    const float bc2 = 1.0f - b2t;
    hipLaunchKernelGGL(sinkhorn_row, dim3(BATCH * 64), dim3(128), XBUF_BYTES, stream,
                       la, log_a, u, v, m_u, v_u, bc1, bc2);
    hipLaunchKernelGGL(sinkhorn_col, dim3(BATCH * 64), dim3(128), XBUF_BYTES, stream,
                       la, log_b, u, v, m_v, v_v, bc1, bc2);
  }
  hipLaunchKernelGGL(sinkhorn_out, dim3((BATCH * NDIM * MDIM) / (4 * 256)), dim3(256),
                     0, stream, la, u, v, (float*)d_out);
}